// QFCModel_65481071408921
// MI455X (gfx1250) — compile-verified
//
#include <hip/hip_runtime.h>
#include <math.h>

typedef float v2f __attribute__((ext_vector_type(2)));
typedef float v8f __attribute__((ext_vector_type(8)));

// D(16x16) = A(16x4) * B(4x16) + C, fp32, wave32.
__device__ __forceinline__ v8f wmma4(v2f a, v2f b, v8f c) {
  return __builtin_amdgcn_wmma_f32_16x16x4_f32(false, a, false, b, (short)0, c,
                                               false, false);
}

// ---------------------------------------------------------------------------
// Kernel 0: build the 16x16 complex unitary of the variational layers.
// Thread t propagates basis state e_t through 3 layers of (Rot x4, CNOT ladder).
// qubit w maps to bit (3-w) of the flattened index (axis i is most significant).
// ---------------------------------------------------------------------------
__global__ void build_unitary(const float* __restrict__ qp,
                              float* __restrict__ Ur, float* __restrict__ Ui) {
  __shared__ float re[16 * 16];
  __shared__ float im[16 * 16];
  int t = threadIdx.x;
  if (t < 16) {
    for (int j = 0; j < 16; ++j) {
      re[t * 16 + j] = (j == t) ? 1.0f : 0.0f;
      im[t * 16 + j] = 0.0f;
    }
    for (int layer = 0; layer < 3; ++layer) {
      for (int wq = 0; wq < 4; ++wq) {
        const float* p = qp + (layer * 4 + wq) * 3;
        float phi = p[0], th = p[1], om = p[2];
        float ch = cosf(0.5f * th), sh = sinf(0.5f * th);
        float ap = 0.5f * (phi + om), am = 0.5f * (phi - om);
        // Rot = RZ(om) RY(th) RZ(phi)
        float u00r = cosf(ap) * ch,  u00i = -sinf(ap) * ch;
        float u01r = -cosf(am) * sh, u01i = -sinf(am) * sh;
        float u10r = cosf(am) * sh,  u10i = -sinf(am) * sh;
        float u11r = cosf(ap) * ch,  u11i = sinf(ap) * ch;
        int mask = 1 << (3 - wq);
        for (int j = 0; j < 16; ++j) {
          if (j & mask) continue;
          int j1 = j | mask;
          float a0r = re[t * 16 + j],  a0i = im[t * 16 + j];
          float a1r = re[t * 16 + j1], a1i = im[t * 16 + j1];
          re[t * 16 + j]  = u00r * a0r - u00i * a0i + u01r * a1r - u01i * a1i;
          im[t * 16 + j]  = u00r * a0i + u00i * a0r + u01r * a1i + u01i * a1r;
          re[t * 16 + j1] = u10r * a0r - u10i * a0i + u11r * a1r - u11i * a1i;
          im[t * 16 + j1] = u10r * a0i + u10i * a0r + u11r * a1i + u11i * a1r;
        }
      }
      for (int cw = 0; cw < 3; ++cw) {
        int cm = 1 << (3 - cw), tm = 1 << (3 - (cw + 1));
        for (int j = 0; j < 16; ++j) {
          if ((j & cm) && !(j & tm)) {
            int j1 = j | tm;
            float tr = re[t * 16 + j]; re[t * 16 + j] = re[t * 16 + j1]; re[t * 16 + j1] = tr;
            float ti = im[t * 16 + j]; im[t * 16 + j] = im[t * 16 + j1]; im[t * 16 + j1] = ti;
          }
        }
      }
    }
    // Column t of U is the final state: U[n][t]
    for (int n = 0; n < 16; ++n) {
      Ur[n * 16 + t] = re[t * 16 + n];
      Ui[n * 16 + t] = im[t * 16 + n];
    }
  }
}

__global__ void init_bn(float* __restrict__ bn) {
  if (threadIdx.x < 8) bn[threadIdx.x] = 0.0f;
}

// ---------------------------------------------------------------------------
// Kernel 2 (fused): pool -> encode -> RX product state -> WMMA unitary apply
//                   -> probs -> Z expvals -> classifier -> BN partial sums.
// Block: 256 threads = 8 wave32; 256 samples per block (32 per wave).
// ---------------------------------------------------------------------------
__global__ void __launch_bounds__(256)
fused(const float* __restrict__ x, const float* __restrict__ enc_w,
      const float* __restrict__ enc_b, const float* __restrict__ cls_w,
      const float* __restrict__ cls_b, const float* __restrict__ Ur,
      const float* __restrict__ Ui, float* __restrict__ out_pre,
      float* __restrict__ bn_acc) {
  __shared__ float lds_bins[8 * 32 * 16];  // pooled sums; reused as probs buffer
  __shared__ float lds_re[8 * 32 * 16];
  __shared__ float lds_im[8 * 32 * 16];
  __shared__ float lds_bn[8];

  int tid = threadIdx.x;
  int wv = tid >> 5;
  int lane = tid & 31;
  int waveSampleBase = blockIdx.x * 256 + wv * 32;

  if (tid < 8) lds_bn[tid] = 0.0f;
  for (int i = tid; i < 8 * 32 * 16; i += 256) lds_bins[i] = 0.0f;
  __syncthreads();

  // --- Phase 1: 6x6 average pooling of x[:, :24, :24] (sum here, /36 later).
  // Each lane pre-reduces a 6-row band in a register -> 1 LDS atomic per band
  // (96 atomics/sample instead of 576), hidden under the coalesced HBM stream.
  if (lane < 24) {
    int bc = lane / 6;  // pooled column
    for (int s = 0; s < 32; ++s) {
      const float* xp = x + (size_t)(waveSampleBase + s) * 784;
      #pragma unroll
      for (int bb = 0; bb < 4; ++bb) {
        float acc = 0.0f;
        #pragma unroll
        for (int rr = 0; rr < 6; ++rr) acc += xp[(bb * 6 + rr) * 28 + lane];
        atomicAdd(&lds_bins[(wv * 32 + s) * 16 + bb * 4 + bc], acc);
      }
    }
  }
  __syncthreads();

  // --- Phase 2: encoder linear + closed-form RX product state (1 lane = 1 sample).
  {
    int s = lane;
    float pooled[16];
    #pragma unroll
    for (int k = 0; k < 16; ++k)
      pooled[k] = lds_bins[(wv * 32 + s) * 16 + k] * (1.0f / 36.0f);
    float cq[4], sq[4];
    #pragma unroll
    for (int q = 0; q < 4; ++q) {
      float z = enc_b[q];
      #pragma unroll
      for (int k = 0; k < 16; ++k) z += enc_w[q * 16 + k] * pooled[k];
      cq[q] = cosf(0.5f * z);
      sq[q] = sinf(0.5f * z);
    }
    // amplitude(j) = Prod_w (bit? sin : cos) * (-i)^popcount(j)
    #pragma unroll
    for (int j = 0; j < 16; ++j) {
      float mag = 1.0f;
      #pragma unroll
      for (int q = 0; q < 4; ++q)
        mag *= ((j >> (3 - q)) & 1) ? sq[q] : cq[q];
      int pc = __popc(j) & 3;
      float rr = (pc == 0) ? mag : ((pc == 2) ? -mag : 0.0f);
      float ii = (pc == 1) ? -mag : ((pc == 3) ? mag : 0.0f);
      lds_re[(wv * 32 + s) * 16 + j] = rr;
      lds_im[(wv * 32 + s) * 16 + j] = ii;
    }
  }
  __syncthreads();

  // --- Phase 3: per wave, two 16-sample tiles through the complex unitary.
  int m = lane & 15;   // A: sample row / B,D: column index n
  int hi = lane >> 4;  // K-group select within a VGPR pair
  for (int h = 0; h < 2; ++h) {
    int sbase = h * 16;
    // A (16x4 f32 layout): lane[m,hi], VGPR v holds K = 4q + v + 2*hi
    v2f are[4], aim[4];
    #pragma unroll
    for (int q = 0; q < 4; ++q) {
      int base = (wv * 32 + sbase + m) * 16 + 4 * q + 2 * hi;
      are[q] = (v2f){lds_re[base], lds_re[base + 1]};
      aim[q] = (v2f){lds_im[base], lds_im[base + 1]};
    }
    // B (4x16 f32 layout): B[k][n] = U[n][k]; VGPR v holds K = 4q + v + 2*hi
    v2f bur[4], bui[4], bnui[4];
    #pragma unroll
    for (int q = 0; q < 4; ++q) {
      #pragma unroll
      for (int v = 0; v < 2; ++v) {
        int k = 4 * q + v + 2 * hi;
        float wr = Ur[m * 16 + k];
        float wi = Ui[m * 16 + k];
        bur[q][v] = wr;
        bui[q][v] = wi;
        bnui[q][v] = -wi;
      }
    }
    // Complex matmul: out = S @ U^T done as 4 real 16x16x16 products.
    v8f dre = {};
    v8f dim_ = {};
    #pragma unroll
    for (int q = 0; q < 4; ++q) dre = wmma4(are[q], bur[q], dre);
    #pragma unroll
    for (int q = 0; q < 4; ++q) dre = wmma4(aim[q], bnui[q], dre);
    #pragma unroll
    for (int q = 0; q < 4; ++q) dim_ = wmma4(are[q], bui[q], dim_);
    #pragma unroll
    for (int q = 0; q < 4; ++q) dim_ = wmma4(aim[q], bur[q], dim_);

    // probs in D layout: VGPR r holds (M = r + 8*hi, N = m)
    #pragma unroll
    for (int r = 0; r < 8; ++r) {
      float p = dre[r] * dre[r] + dim_[r] * dim_[r];
      lds_bins[wv * 512 + (r + 8 * hi) * 16 + m] = p;
    }
    __syncthreads();  // uniform; also orders probs store -> load

    if (lane < 16) {
      float p[16];
      #pragma unroll
      for (int n = 0; n < 16; ++n) p[n] = lds_bins[wv * 512 + lane * 16 + n];
      float ev[4];
      #pragma unroll
      for (int q = 0; q < 4; ++q) {
        float e = 0.0f;
        #pragma unroll
        for (int n = 0; n < 16; ++n)
          e += (((n >> (3 - q)) & 1) ? -p[n] : p[n]);
        ev[q] = e;
      }
      int g = waveSampleBase + sbase + lane;
      #pragma unroll
      for (int c = 0; c < 4; ++c) {
        float o = cls_b[c];
        #pragma unroll
        for (int q = 0; q < 4; ++q) o += cls_w[c * 4 + q] * ev[q];
        out_pre[g * 4 + c] = o;
        atomicAdd(&lds_bn[c], o);
        atomicAdd(&lds_bn[4 + c], o * o);
      }
    }
    __syncthreads();
  }
  if (tid < 8) atomicAdd(&bn_acc[tid], lds_bn[tid]);
}

// ---------------------------------------------------------------------------
// Kernel 3: BatchNorm1d (training mode, biased stats, eps=1e-5).
// ---------------------------------------------------------------------------
__global__ void bn_finalize(const float* __restrict__ out_pre,
                            const float* __restrict__ bn_acc,
                            const float* __restrict__ gamma,
                            const float* __restrict__ beta,
                            float* __restrict__ out, int total, float invB) {
  int i = blockIdx.x * blockDim.x + threadIdx.x;
  if (i < total) {
    int c = i & 3;
    float mean = bn_acc[c] * invB;
    float var = bn_acc[4 + c] * invB - mean * mean;
    out[i] = (out_pre[i] - mean) * rsqrtf(var + 1e-5f) * gamma[c] + beta[c];
  }
}

extern "C" void kernel_launch(void* const* d_in, const int* in_sizes, int n_in,
                              void* d_out, int out_size, void* d_ws,
                              size_t ws_size, hipStream_t stream) {
  const float* x       = (const float*)d_in[0];
  const float* enc_w   = (const float*)d_in[1];
  const float* enc_b   = (const float*)d_in[2];
  const float* qparams = (const float*)d_in[3];
  const float* cls_w   = (const float*)d_in[4];
  const float* cls_b   = (const float*)d_in[5];
  const float* bn_g    = (const float*)d_in[6];
  const float* bn_b    = (const float*)d_in[7];

  int B = in_sizes[0] / 784;

  float* ws = (float*)d_ws;
  float* Ur      = ws;          // 256
  float* Ui      = ws + 256;    // 256
  float* bn      = ws + 512;    // 8 (sum[4], sumsq[4])
  float* out_pre = ws + 1024;   // B*4

  build_unitary<<<1, 32, 0, stream>>>(qparams, Ur, Ui);
  init_bn<<<1, 32, 0, stream>>>(bn);
  fused<<<B / 256, 256, 0, stream>>>(x, enc_w, enc_b, cls_w, cls_b, Ur, Ui,
                                     out_pre, bn);
  int total = B * 4;
  bn_finalize<<<(total + 255) / 256, 256, 0, stream>>>(
      out_pre, bn, bn_g, bn_b, (float*)d_out, total, 1.0f / (float)B);
}